// HeteroSoap_79757542687234
// MI455X (gfx1250) — compile-verified
//
#include <hip/hip_runtime.h>
#include <hip/hip_bf16.h>

typedef float v2f __attribute__((ext_vector_type(2)));
typedef float v8f __attribute__((ext_vector_type(8)));

namespace {
constexpr int   kWavesPerBlock = 8;
constexpr int   kBlock         = 256;   // 8 wave32s
constexpr int   kGrid          = 512;
constexpr int   kStr           = 33;    // per-atom LDS stride in floats (16 u + 16 Y + pad)
constexpr float kUnit          = 2.0f;  // RC / 3
constexpr float kRc            = 6.0f;
}

// ---------------------------------------------------------------------------
// Stage 0: zero the 256-float c accumulator in workspace
// ---------------------------------------------------------------------------
__global__ void soap_zero_c(float* __restrict__ c) {
  c[threadIdx.x] = 0.0f;
}

// ---------------------------------------------------------------------------
// Stage 1: streaming reduction  c[s,k,a,b] = sum_atoms onehot[s]*f[k]*Y[a,b]
// mapped onto V_WMMA_F32_16X16X4_F32 rank-4 updates (4 atoms per WMMA).
//   M = s*4+k  (A-matrix rows = u vector)
//   N = a*4+b  (B-matrix cols = packed solid harmonics Y)
//   K = atom slot within group of 4
// ---------------------------------------------------------------------------
__global__ __launch_bounds__(kBlock) void soap_accum(
    const float* __restrict__ coo, const int* __restrict__ numbers,
    float* __restrict__ c_out, int n_atoms)
{
  __shared__ float lds[kWavesPerBlock * 32 * kStr];

  const int lane  = threadIdx.x & 31;
  const int wave  = threadIdx.x >> 5;
  const int mlane = lane & 15;   // matrix row/col index within half-wave
  const int hi    = lane >> 4;   // half-wave select (K slot 0/1 vs 2/3)
  float* wl = &lds[wave * 32 * kStr];

  v8f acc = {};                  // persistent 16x16 f32 accumulator (8 VGPRs)

  const int chunks  = (n_atoms + 31) >> 5;
  const int wstride = gridDim.x * kWavesPerBlock;
  const int w0      = blockIdx.x * kWavesPerBlock + wave;

  for (int ch = w0; ch < chunks; ch += wstride) {
    const int atom = ch * 32 + lane;

    // ---- per-lane atom compute ------------------------------------------
    float x = 0.0f, y = 0.0f, z = 0.0f;
    int spec = -1;                       // invalid species -> u == 0
    if (atom < n_atoms) {
      x = coo[atom * 3 + 0] * (1.0f / kUnit);
      y = coo[atom * 3 + 1] * (1.0f / kUnit);
      z = coo[atom * 3 + 2] * (1.0f / kUnit);
      spec = numbers[atom];
    }
    const float r2   = x * x + y * y + z * z;
    const float dist = kUnit * sqrtf(r2);
    float t   = (dist < kRc) ? (1.0f - dist * (1.0f / kRc)) : 0.0f;
    const float rad = __expf(-0.5f * r2) * t * t;

    const float f0 = rad;
    const float f1 = rad * r2;
    const float f2 = f1 * r2;
    const float f3 = f2 * r2;
    const float fk[4] = {f0, f1, f2, f3};

    // ---- solid harmonics recursion up to l = 3 --------------------------
    const float re00 = 0.28209479177387814f;          // 1/sqrt(4*pi)
    const float cl1 = -1.2247448713915890f;           // -sqrt(3/2)
    const float g1  =  1.7320508075688772f;           //  sqrt(3)
    const float re11 = cl1 * x * re00;
    const float im11 = cl1 * y * re00;
    const float re10 = g1 * z * re00;

    const float cl2 = -1.1180339887498949f;           // -sqrt(5/4)
    const float g2  =  2.2360679774997896f;           //  sqrt(5)
    const float re22 = cl2 * (x * re11 - y * im11);
    const float im22 = cl2 * (x * im11 + y * re11);
    const float re21 = g2 * z * re11;
    const float im21 = g2 * z * im11;
    const float a20 = 1.9364916731037085f;            // sqrt(15/4)
    const float b20 = 0.57735026918962576f;           // sqrt(1/3)
    const float re20 = a20 * (z * re10 - b20 * r2 * re00);

    const float cl3 = -1.0801234497346435f;           // -sqrt(7/6)
    const float g3  =  2.6457513110645906f;           //  sqrt(7)
    const float re33 = cl3 * (x * re22 - y * im22);
    const float im33 = cl3 * (x * im22 + y * re22);
    const float re32 = g3 * z * re22;
    const float im32 = g3 * z * im22;
    const float a30 = 1.9720265943665387f;            // sqrt(35/9)
    const float b30 = 0.51639777949432225f;           // sqrt(4/15)
    const float re30 = a30 * (z * re20 - b30 * r2 * re10);
    const float a31 = 2.0916500663351889f;            // sqrt(35/8)
    const float b31 = 0.44721359549995793f;           // sqrt(1/5)
    const float re31 = a31 * (z * re21 - b31 * r2 * re11);
    const float im31 = a31 * (z * im21 - b31 * r2 * im11);

    // packed Y: Y[l, l-m] = Re, Y[l-m, l] = Im  (row-major 4x4)
    const float Yv[16] = {
        re00, im11, im22, im33,
        re11, re10, im21, im32,
        re22, re21, re20, im31,
        re33, re32, re31, re30};

    // ---- stage to LDS: u[16] then Y[16] per atom ------------------------
    #pragma unroll
    for (int m = 0; m < 16; ++m) {
      wl[lane * kStr + m]      = ((m >> 2) == spec) ? fk[m & 3] : 0.0f;
      wl[lane * kStr + 16 + m] = Yv[m];
    }
    __builtin_amdgcn_wave_barrier();
    asm volatile("s_wait_dscnt 0x0" ::: "memory");   // same-wave LDS RAW

    // ---- 8 x V_WMMA_F32_16X16X4_F32: 4 atoms each -----------------------
    // A/B f32 layout: VGPR0 holds K{0,2}, VGPR1 holds K{1,3};
    // lanes 0-15 = low K of pair, lanes 16-31 = high K of pair.
    #pragma unroll
    for (int g = 0; g < 8; ++g) {
      const int sel0 = g * 4 + (hi ? 2 : 0);
      const int sel1 = g * 4 + (hi ? 3 : 1);
      v2f A, B;
      A.x = wl[sel0 * kStr + mlane];        // u of atom in K slot 0/2
      A.y = wl[sel1 * kStr + mlane];        // u of atom in K slot 1/3
      B.x = wl[sel0 * kStr + 16 + mlane];   // Y of atom in K slot 0/2
      B.y = wl[sel1 * kStr + 16 + mlane];   // Y of atom in K slot 1/3
      acc = __builtin_amdgcn_wmma_f32_16x16x4_f32(
          /*neg_a=*/false, A, /*neg_b=*/false, B,
          /*c_mod=*/(short)0, acc, /*reuse_a=*/false, /*reuse_b=*/false);
    }
    __builtin_amdgcn_wave_barrier();
    asm volatile("s_wait_dscnt 0x0" ::: "memory");   // loads done before reuse
  }

  // ---- merge: C VGPR j holds (M=j, N=lane) / (M=j+8, N=lane-16) ---------
  #pragma unroll
  for (int j = 0; j < 8; ++j) {
    const int M = j + 8 * hi;
    const int N = mlane;
    unsafeAtomicAdd(&c_out[M * 16 + N], acc[j]);
  }
}

// ---------------------------------------------------------------------------
// Stage 2: tiny einsum  p[a,b,n,m,l] = (Σ_j c[b,m,l,j]c[a,n,l,j]Yr[l,j]
//                                     + Σ_i c[b,m,i,l]c[a,n,i,l]Yi[i,l]) * nnl
// ---------------------------------------------------------------------------
__device__ __forceinline__ float soap_a_nl(int n, int l) {
  const float fact[7] = {1.f, 1.f, 2.f, 6.f, 24.f, 120.f, 720.f};
  return 1.0f / ((float)(2 * l + 1) * exp2f((float)(2 * n + l)) *
                 fact[n] * fact[n + l]);
}

__global__ void soap_finalize(const float* __restrict__ c,
                              float* __restrict__ out) {
  const int idx = blockIdx.x * blockDim.x + threadIdx.x;
  if (idx >= 1024) return;
  const int l = idx & 3;
  const int m = (idx >> 2) & 3;
  const int n = (idx >> 4) & 3;
  const int b = (idx >> 6) & 3;
  const int a = (idx >> 8) & 3;

  float p = 0.0f;
  // Yr[l][j] = 1 on diag, 2 strictly below
  for (int j = 0; j <= l; ++j) {
    const float wgt = (j == l) ? 1.0f : 2.0f;
    p += wgt * c[((b * 4 + m) * 4 + l) * 4 + j] *
               c[((a * 4 + n) * 4 + l) * 4 + j];
  }
  // Yi[i][l] = 2 strictly above diag (i < l)
  for (int i = 0; i < l; ++i) {
    p += 2.0f * c[((b * 4 + m) * 4 + i) * 4 + l] *
                c[((a * 4 + n) * 4 + i) * 4 + l];
  }
  p *= sqrtf(soap_a_nl(n, l) * soap_a_nl(m, l));
  out[idx] = p;
}

// ---------------------------------------------------------------------------
extern "C" void kernel_launch(void* const* d_in, const int* in_sizes, int n_in,
                              void* d_out, int out_size, void* d_ws,
                              size_t ws_size, hipStream_t stream) {
  const float* coo     = (const float*)d_in[0];
  const int*   numbers = (const int*)d_in[1];
  float*       out     = (float*)d_out;
  float*       c_ws    = (float*)d_ws;   // 256 floats
  const int n_atoms = in_sizes[0] / 3;

  soap_zero_c<<<1, 256, 0, stream>>>(c_ws);
  soap_accum<<<kGrid, kBlock, 0, stream>>>(coo, numbers, c_ws, n_atoms);
  soap_finalize<<<4, 256, 0, stream>>>(c_ws, out);
}